// DLADMMNet_46033459478794
// MI455X (gfx1250) — compile-verified
//
#include <hip/hip_runtime.h>

typedef __attribute__((ext_vector_type(2))) float v2f;
typedef __attribute__((ext_vector_type(8))) float v8f;

#define M_DIM   512
#define N_B     4096
#define D_DIM   1024
#define NLAYERS 16

#define TILE       16
#define ROWS       32          // M rows per block (2 WMMA tiles per wave)
#define KC         128         // K chunk staged in LDS per buffer
#define LDS_STRIDE (KC + 4)    // padded row stride (floats) -> conflict-free b64 reads
#define BLOCK_COLS 128         // 8 waves x 16 cols
#define NTHREADS   256
#define PF         (ROWS / 8)  // async-load instructions per wave per chunk (one row each)

#define LDS_FLOATS (2 * ROWS * LDS_STRIDE)   // double buffer

__device__ __forceinline__ float softshrink(float v, float t) {
    float s = fmaxf(fabsf(v) - t, 0.0f);
    return copysignf(s, v);
}

// Async global -> LDS copy of one [ROWS x KC] chunk (double-buffered).
// Wave w, lane l, step i moves row (i*8 + w), cols [4l, 4l+4) : 512B/wave/instr.
__device__ __forceinline__ void prefetch_chunk(const float* __restrict__ ablk, int ldk,
                                               unsigned ldsbase_f, int wave, int lane) {
    #pragma unroll
    for (int i = 0; i < PF; ++i) {
        const int row = i * 8 + wave;
        const int col = lane * 4;
        const unsigned lds_off = (ldsbase_f + (unsigned)(row * LDS_STRIDE + col)) * 4u;
        const float* g = ablk + (size_t)row * ldk + col;
        asm volatile("global_load_async_to_lds_b128 %0, %1, off"
                     :: "v"(lds_off), "v"(g)
                     : "memory");
    }
}

template<int N>
__device__ __forceinline__ void wait_async() {
    asm volatile("s_wait_asynccnt %0" :: "n"(N) : "memory");
}

// MODE 0: step GEMM1  (AZ -> En, Ln, Varn)
// MODE 1: init GEMM1  (AZ -> Var0; copy E0->E_all[0], L0->L_all[0])
// MODE 2: GEMM2       (W@Var -> Zn = soft(Zp - W@Var, ap))
template<int K, int MODE>
__global__ __launch_bounds__(NTHREADS)
void dladmm_fused(const float* __restrict__ Amat,   // rows x K, row-major (ld == K)
                  const float* __restrict__ Bmat,   // K x N_B, row-major
                  const float* __restrict__ Ep,
                  const float* __restrict__ Lp,
                  const float* __restrict__ xin,    // MODE2: Zp
                  const float* __restrict__ p0,     // b1 / b1 / ap
                  const float* __restrict__ p1,     // b2
                  const float* __restrict__ p2,     // b3
                  const float* __restrict__ p3,     // s2
                  const float* __restrict__ p4,     // ap1
                  float* __restrict__ out0,         // Var_ws / Var_ws / Z_out
                  float* __restrict__ out1,         // E_out
                  float* __restrict__ out2)         // L_out
{
    extern __shared__ float smem[];   // LDS_FLOATS floats, base offset 0

    const int lane = threadIdx.x & 31;
    const int wave = threadIdx.x >> 5;
    const int mb   = blockIdx.y * ROWS;                       // row-block base
    const int nb   = blockIdx.x * BLOCK_COLS + wave * TILE;   // col-tile base

    const int half = lane >> 4;   // 0: lanes 0-15, 1: lanes 16-31
    const int l16  = lane & 15;

    constexpr int NCHUNK = K / KC;

    v8f acc0 = {0.f, 0.f, 0.f, 0.f, 0.f, 0.f, 0.f, 0.f};
    v8f acc1 = acc0;

    const float* ablk = Amat + (size_t)mb * K;

    // Pipeline: prefetch chunk 0, then per chunk prefetch next while computing.
    prefetch_chunk(ablk, K, 0u, wave, lane);

    for (int c = 0; c < NCHUNK; ++c) {
        const unsigned curbase = (unsigned)((c & 1) * ROWS * LDS_STRIDE);
        if (c + 1 < NCHUNK) {
            const unsigned nxtbase = (unsigned)(((c + 1) & 1) * ROWS * LDS_STRIDE);
            prefetch_chunk(ablk + (size_t)(c + 1) * KC, K, nxtbase, wave, lane);
            wait_async<PF>();     // chunk c's loads complete (in-order per wave)
        } else {
            wait_async<0>();
        }
        __syncthreads();          // all waves' chunk-c data visible in LDS

        const float* abuf = smem + curbase;
        const float* bc   = Bmat + (size_t)c * KC * N_B + nb + l16;

        #pragma unroll 4
        for (int k = 0; k < KC; k += 4) {
            const int krow = k + 2 * half;  // lanes 0-15: K=k,k+1 ; 16-31: K=k+2,k+3
            v2f a0 = *reinterpret_cast<const v2f*>(&abuf[l16 * LDS_STRIDE + krow]);
            v2f a1 = *reinterpret_cast<const v2f*>(&abuf[(16 + l16) * LDS_STRIDE + krow]);
            v2f b;
            b.x = bc[(size_t)krow * N_B];
            b.y = bc[(size_t)(krow + 1) * N_B];
            acc0 = __builtin_amdgcn_wmma_f32_16x16x4_f32(
                       false, a0, false, b, (short)0, acc0, false, false);
            acc1 = __builtin_amdgcn_wmma_f32_16x16x4_f32(
                       false, a1, false, b, (short)0, acc1, false, false);
        }
        __syncthreads();          // done reading buf before it is overwritten
    }

    // Epilogue: tile t, vgpr r  ->  row = mb + 16*t + r + 8*half, col = nb + l16
    const int col = nb + l16;
    #pragma unroll
    for (int t = 0; t < 2; ++t) {
        #pragma unroll
        for (int r = 0; r < 8; ++r) {
            const int row = mb + 16 * t + r + 8 * half;
            const size_t idx = (size_t)row * N_B + col;
            const float az = (t == 0) ? acc0[r] : acc1[r];
            if (MODE == 2) {
                const float zp = xin[idx];
                const float ap = p0[0];
                out0[idx] = softshrink(zp - az, ap);
            } else if (MODE == 1) {
                const float ep = Ep[idx], lp = Lp[idx], xv = xin[idx];
                const float b1 = p0[0];
                const float t0 = az + ep - xv;
                out0[idx] = lp + b1 * t0;   // Var0
                out1[idx] = ep;             // E_all[0] = E0
                out2[idx] = lp;             // L_all[0] = L0
            } else {
                const float ep = Ep[idx], lp = Lp[idx], xv = xin[idx];
                const float b1 = p0[0], b2 = p1[0], b3 = p2[0];
                const float s2 = p3[0], ap1 = p4[0];
                const float t0 = az + ep - xv;
                const float vv = lp + b2 * t0;
                const float en = softshrink(ep - s2 * vv, ap1);
                const float tn = az + en - xv;
                const float ln = lp + b3 * tn;
                out1[idx] = en;
                out2[idx] = ln;
                out0[idx] = ln + b1 * tn;   // Varn
            }
        }
    }
}

extern "C" void kernel_launch(void* const* d_in, const int* in_sizes, int n_in,
                              void* d_out, int out_size, void* d_ws, size_t ws_size,
                              hipStream_t stream) {
    const float* x     = (const float*)d_in[0];
    const float* A     = (const float*)d_in[1];
    const float* W     = (const float*)d_in[2];
    const float* Z0    = (const float*)d_in[3];
    const float* E0    = (const float*)d_in[4];
    const float* L0    = (const float*)d_in[5];
    const float* beta1 = (const float*)d_in[6];
    const float* beta2 = (const float*)d_in[7];
    const float* beta3 = (const float*)d_in[8];
    const float* ss2   = (const float*)d_in[9];
    const float* apar  = (const float*)d_in[10];
    const float* apar1 = (const float*)d_in[11];

    float* Zall = (float*)d_out;                                  // 16 x 1024 x 4096
    float* Eall = Zall + (size_t)NLAYERS * D_DIM * N_B;           // 16 x 512 x 4096
    float* Lall = Eall + (size_t)NLAYERS * M_DIM * N_B;           // 16 x 512 x 4096
    float* Var  = (float*)d_ws;                                   // 512 x 4096

    const size_t shbytes = (size_t)LDS_FLOATS * sizeof(float);

    dim3 block(NTHREADS);
    dim3 g1(N_B / BLOCK_COLS, M_DIM / ROWS);   // 32 x 16
    dim3 g2(N_B / BLOCK_COLS, D_DIM / ROWS);   // 32 x 32

    // ----- init layer: Var0 = L0 + beta1[0]*(A@Z0 + E0 - x); Z1 = soft(Z0 - W0@Var0) -----
    dladmm_fused<D_DIM, 1><<<g1, block, shbytes, stream>>>(
        A, Z0, E0, L0, x,
        beta1, nullptr, nullptr, nullptr, nullptr,
        Var, Eall, Lall);
    dladmm_fused<M_DIM, 2><<<g2, block, shbytes, stream>>>(
        W, Var, nullptr, nullptr, Z0,
        apar, nullptr, nullptr, nullptr, nullptr,
        Zall, nullptr, nullptr);

    // ----- layers 1..15 -----
    for (int i = 1; i < NLAYERS; ++i) {
        const float* Zp  = Zall + (size_t)(i - 1) * D_DIM * N_B;
        const float* Epr = Eall + (size_t)(i - 1) * M_DIM * N_B;
        const float* Lpr = Lall + (size_t)(i - 1) * M_DIM * N_B;

        dladmm_fused<D_DIM, 0><<<g1, block, shbytes, stream>>>(
            A, Zp, Epr, Lpr, x,
            beta1 + i, beta2 + (i - 1), beta3 + (i - 1), ss2 + (i - 1), apar1 + (i - 1),
            Var,
            Eall + (size_t)i * M_DIM * N_B,
            Lall + (size_t)i * M_DIM * N_B);

        dladmm_fused<M_DIM, 2><<<g2, block, shbytes, stream>>>(
            W + (size_t)i * D_DIM * M_DIM, Var, nullptr, nullptr, Zp,
            apar + i, nullptr, nullptr, nullptr, nullptr,
            Zall + (size_t)i * D_DIM * N_B, nullptr, nullptr);
    }
}